// BulkHamiltonian_36146444764071
// MI455X (gfx1250) — compile-verified
//
#include <hip/hip_runtime.h>

#if defined(__HIP_DEVICE_COMPILE__) && __has_builtin(__builtin_amdgcn_tensor_store_from_lds)
#define HAVE_TDM 1
#else
#define HAVE_TDM 0
#endif

typedef __attribute__((ext_vector_type(4))) unsigned int v4u;
typedef __attribute__((ext_vector_type(8))) int v8i;
typedef __attribute__((ext_vector_type(4))) int v4i;

#define TPB 256
#define PTS 256               // k-points per tile (one per thread)
#define SLOT_F 128            // floats per point (8x8 complex64)
#define TILE_F (PTS * SLOT_F) // floats per LDS buffer
#define TILE_BYTES (TILE_F * 4) // 128 KB per buffer, double buffered

// Variable part: the four float4 chunks holding -A_tr (rows 4-5, cols 2-3)
// and -A_bl = conj(-A_tr) (rows 6-7, cols 0-1).
__device__ __forceinline__ void compute_var(float kx, float ky,
                                            float4& r4, float4& r5,
                                            float4& r6, float4& r7) {
  const float SQ3 = 1.7320508075688772f;
  const float Q   = 0.43301270189221932f; // sqrt(3)/4
  float p1 = kx * SQ3;                      // k . a1
  float p2 = kx * (SQ3 * 0.5f) + ky * 1.5f; // k . a2
  float s1, c1, s2, c2;
  sincosf(p1, &s1, &c1);
  sincosf(p2, &s2, &c2);
  // T = -A_tr (uses conj(K1), conj(K2)); B = -A_bl = conj(T)
  float t00r = -0.75f * (1.f + c1);
  float t00i =  0.75f * s1;
  float t01r =  Q * (c1 - 1.f);
  float t01i = -Q * s1;
  float t11r = -0.25f - 0.25f * c1 - c2;
  float t11i =  0.25f * s1 + s2;
  r4 = make_float4(t00r,  t00i, t01r,  t01i); // row4 cols2-3
  r5 = make_float4(t01r,  t01i, t11r,  t11i); // row5 cols2-3
  r6 = make_float4(t00r, -t00i, t01r, -t01i); // row6 cols0-1
  r7 = make_float4(t01r, -t01i, t11r, -t11i); // row7 cols0-1
}

// Constant part of one 128-float H slot (everything except the 4 variable chunks).
__device__ __forceinline__ void fill_const_slot(float* s) {
  float4 z = make_float4(0.f, 0.f, 0.f, 0.f);
  float4* v = (float4*)s;
#pragma unroll
  for (int i = 0; i < 32; ++i) v[i] = z;
  // rows 0-3: [0 | I]
  s[8]  = 1.0f; s[26] = 1.0f; s[44] = 1.0f; s[62] = 1.0f;
  // row4: R00=1.5 ; L12 row0 col5 = +0.2i
  s[64] = 1.5f; s[75] = 0.2f;
  // row5: R11=1.5 ; L12 row1 col4 = -0.2i
  s[82] = 1.5f; s[89] = -0.2f;
  // row6: R00=1.5 (cols2-3) ; L12 row2 col7 = +0.2i
  s[100] = 1.5f; s[111] = 0.2f;
  // row7: R11=1.5 ; L12 row3 col4 = -0.2i
  s[118] = 1.5f; s[121] = -0.2f;
}

__global__ void __launch_bounds__(TPB)
ham_tdm(const float* __restrict__ k, float* __restrict__ out,
        int B, int num_tiles) {
#if HAVE_TDM
  extern __shared__ float smem[];
  // Fill constants of both double buffers once; TDM only reads them afterwards.
  for (int buf = 0; buf < 2; ++buf)
    fill_const_slot(smem + buf * TILE_F + threadIdx.x * SLOT_F);
  __syncthreads();

  int it = 0;
  for (int tile = blockIdx.x; tile < num_tiles; tile += gridDim.x, ++it) {
    int buf = it & 1;
    // Before overwriting this buffer, the TDM store issued 2 iterations ago
    // (same buffer) must be done; allow the previous iteration's store to fly.
    if (threadIdx.x == 0) __builtin_amdgcn_s_wait_tensorcnt(1);
    __syncthreads();

    int b0 = tile * PTS;
    int npts = B - b0; if (npts > PTS) npts = PTS;
    if ((int)threadIdx.x < npts) {
      int b = b0 + (int)threadIdx.x;
      float4 r4, r5, r6, r7;
      compute_var(k[2 * b], k[2 * b + 1], r4, r5, r6, r7);
      float4* slot = (float4*)(smem + buf * TILE_F + threadIdx.x * SLOT_F);
      slot[17] = r4;  // floats 68-71
      slot[21] = r5;  // floats 84-87
      slot[24] = r6;  // floats 96-99
      slot[28] = r7;  // floats 112-115
    }
    __syncthreads();

    if (threadIdx.x == 0) {
      // Flat 1-D TDM store: npts*64 elements of 8 bytes, LDS -> global.
      unsigned lds_addr = (unsigned)(unsigned long long)(smem + buf * TILE_F);
      unsigned long long ga =
          (unsigned long long)out + (unsigned long long)b0 * 512ull;
      unsigned n = (unsigned)npts * 64u; // 8-byte elements, <= 16384

      v4u g0;
      g0.x = 1u;                                   // count=1 (valid), user mode
      g0.y = lds_addr;                             // lds_addr[31:0]
      g0.z = (unsigned)(ga & 0xffffffffu);         // global_addr[31:0]
      g0.w = (unsigned)((ga >> 32) & 0x1ffffffu)   // global_addr[56:32]
             | (2u << 30);                         // type=2 (image/tensor)

      v8i g1;
      g1[0] = (int)(3u << 16);                     // wg_mask=0, data_size=8B
      g1[1] = (int)((n & 0xffffu) << 16);          // abar=0 | tensor_dim0[15:0]
      g1[2] = (int)((n >> 16) & 0xffffu)           // tensor_dim0[31:16]
              | (1 << 16);                         // tensor_dim1 = 1
      g1[3] = (int)((n & 0xffffu) << 16);          // dim1 hi=0 | tile_dim0 = n
      g1[4] = 0;                                   // tile_dim1=0, tile_dim2=0
      g1[5] = (int)n;                              // tensor_dim0_stride
      g1[6] = 0;                                   // stride0 hi | stride1 lo
      g1[7] = 0;

      v4i g2; g2[0] = 1; g2[1] = 1; g2[2] = 0; g2[3] = 0; // dims 2/3 = 1
      v4i g3; g3[0] = 0; g3[1] = (1 << 16); g3[2] = 0; g3[3] = 0; // dim4=1
      v8i g4; g4[0] = 0; g4[1] = 0; g4[2] = 0; g4[3] = 0;
              g4[4] = 0; g4[5] = 0; g4[6] = 0; g4[7] = 0; // VADDR4 unused/NULL

      __builtin_amdgcn_tensor_store_from_lds(g0, g1, g2, g3, g4, 0);
    }
  }
  if (threadIdx.x == 0) __builtin_amdgcn_s_wait_tensorcnt(0);
#else
  // Fallback: direct per-point stores (no TDM builtin available).
  for (int tile = blockIdx.x; tile < num_tiles; tile += gridDim.x) {
    int b0 = tile * PTS;
    int npts = B - b0; if (npts > PTS) npts = PTS;
    if ((int)threadIdx.x < npts) {
      int b = b0 + (int)threadIdx.x;
      float4 r4, r5, r6, r7;
      compute_var(k[2 * b], k[2 * b + 1], r4, r5, r6, r7);
      float s[SLOT_F];
      fill_const_slot(s);
      float4* sv = (float4*)s;
      sv[17] = r4; sv[21] = r5; sv[24] = r6; sv[28] = r7;
      float4* o = (float4*)(out + (unsigned long long)b * SLOT_F);
#pragma unroll
      for (int i = 0; i < 32; ++i) o[i] = sv[i];
    }
  }
#endif
}

extern "C" void kernel_launch(void* const* d_in, const int* in_sizes, int n_in,
                              void* d_out, int out_size, void* d_ws, size_t ws_size,
                              hipStream_t stream) {
  const float* k = (const float*)d_in[0];
  float* out = (float*)d_out;
  int B = in_sizes[0] / 2;                 // k is (B,2,1) float32
  int num_tiles = (B + PTS - 1) / PTS;
  int grid = num_tiles < 512 ? num_tiles : 512;
  // 256 KB dynamic LDS (double-buffered tiles); raise the cap explicitly.
  (void)hipFuncSetAttribute(reinterpret_cast<const void*>(ham_tdm),
                            hipFuncAttributeMaxDynamicSharedMemorySize,
                            2 * TILE_BYTES);
  ham_tdm<<<grid, TPB, 2 * TILE_BYTES, stream>>>(k, out, B, num_tiles);
}